// SymmetricContraction_63977832841983
// MI455X (gfx1250) — compile-verified
//
#include <hip/hip_runtime.h>

// CDNA5 (gfx1250, wave32) symmetric tensor contraction via monomial-GEMM.
//   g[n*m, col] = sum_k Mcoef[k, col] * mono[n*m, k]   (V_WMMA_F32_16X16X4_F32)
//   out[n,m,p]  = sum_col Wpack[sp, col, m] * g[...]
// Mpair = symmetrized basis matrix, pre-interleaved by K-pair and zero-padded
// so each workgroup can DMA it into LDS with ONE Tensor-Data-Mover op.

typedef float v2f __attribute__((ext_vector_type(2)));
typedef float v8f __attribute__((ext_vector_type(8)));
typedef int   v4i __attribute__((ext_vector_type(4)));
typedef int   v8i __attribute__((ext_vector_type(8)));

#define NNODE   2048
#define MUL     128
#define DIN     9
#define KDIM    220    // 9 + 45 + 165 monomials, padded to mult of 4
#define NPAD    32     // 23 g-functions padded to 2 N-tiles
#define NCOL    23
#define WSTR    24     // Wpack col stride
#define GSTR    25     // LDS g-buffer stride (odd -> conflict-free)
#define BLSF    7168   // Mpair floats: 224 k-rows x 32 cols (TDM tile, 28 KB)

// out-component target per g-function column
__device__ __constant__ int kPtgt[NCOL] =
    {0, 1,2,3, 0,0,0, 1,2,3,1,2,3, 0,0,0,0, 1,2,3,1,2,3};

// ---------------- setup kernel 1: symmetrized basis matrix --------------
// Writes directly in the K-pair interleaved LDS layout:
//   Mpair[(k>>1)*64 + col*2 + (k&1)],  k in [0,224)  (rows >= 219 are zero)
__global__ void build_M(const float* __restrict__ b0, const float* __restrict__ b1,
                        const float* __restrict__ b2, const float* __restrict__ b3,
                        const float* __restrict__ b4, const float* __restrict__ b5,
                        float* __restrict__ Mpair) {
  int j = blockIdx.x * blockDim.x + threadIdx.x;
  if (j >= 224) return;
  float cols[NPAD];
  for (int c = 0; c < NPAD; ++c) cols[c] = 0.f;

  if (j < 9) {                                   // degree-1 monomial f[a]
    int a = j;
    cols[0] = b0[a];
    for (int p = 0; p < 3; ++p) cols[1 + p] = b1[a * 3 + p];
  } else if (j < 54) {                           // degree-2 monomial f[a]f[b], a<=b
    int t = j - 9, a = 0, b = 0, cnt = 0;
    for (int aa = 0; aa < 9; ++aa)
      for (int bb = aa; bb < 9; ++bb) { if (cnt == t) { a = aa; b = bb; } ++cnt; }
    for (int s = 0; s < 3; ++s) {
      float v = b2[(a * 9 + b) * 3 + s];
      if (a != b) v += b2[(b * 9 + a) * 3 + s];
      cols[4 + s] = v;
    }
    for (int s = 0; s < 2; ++s)
      for (int p = 0; p < 3; ++p) {
        float v = b3[((a * 9 + b) * 2 + s) * 3 + p];
        if (a != b) v += b3[((b * 9 + a) * 2 + s) * 3 + p];
        cols[7 + s * 3 + p] = v;
      }
  } else if (j < 219) {                          // degree-3 monomial, a<=b<=c
    int t = j - 54, a = 0, b = 0, c = 0, cnt = 0;
    for (int aa = 0; aa < 9; ++aa)
      for (int bb = aa; bb < 9; ++bb)
        for (int cc = bb; cc < 9; ++cc) { if (cnt == t) { a = aa; b = bb; c = cc; } ++cnt; }
    int px[6] = {a, a, b, b, c, c};
    int py[6] = {b, c, a, c, a, b};
    int pz[6] = {c, b, c, a, b, a};
    for (int i = 0; i < 6; ++i) {                // distinct permutations only
      bool dup = false;
      for (int k = 0; k < i; ++k)
        if (px[k] == px[i] && py[k] == py[i] && pz[k] == pz[i]) dup = true;
      if (dup) continue;
      int base = (px[i] * 9 + py[i]) * 9 + pz[i];
      for (int s = 0; s < 4; ++s) cols[13 + s] += b4[base * 4 + s];
      for (int s = 0; s < 2; ++s)
        for (int p = 0; p < 3; ++p) cols[17 + s * 3 + p] += b5[(base * 2 + s) * 3 + p];
    }
  }                                              // j in [219,224): zero pad rows
  for (int c = 0; c < NPAD; ++c)
    Mpair[(j >> 1) * (NPAD * 2) + c * 2 + (j & 1)] = cols[c];
}

// ---------------- setup kernel 2: packed per-(species,col,m) weights ----
__global__ void build_W(const float* __restrict__ w0, const float* __restrict__ w1,
                        const float* __restrict__ w2, const float* __restrict__ w3,
                        const float* __restrict__ w4, const float* __restrict__ w5,
                        float* __restrict__ Wpack) {
  int tid = blockIdx.x * blockDim.x + threadIdx.x;
  if (tid >= 10 * WSTR * MUL) return;
  int sp  = tid / (WSTR * MUL);
  int r   = tid % (WSTR * MUL);
  int col = r / MUL;
  int m   = r % MUL;
  float v = 0.f;
  if      (col == 0)  v = w0[(sp * 1 + 0) * MUL + m];
  else if (col <= 3)  v = w1[(sp * 1 + 0) * MUL + m];
  else if (col <= 6)  v = w2[(sp * 3 + (col - 4)) * MUL + m];
  else if (col <= 12) v = w3[(sp * 2 + (col - 7) / 3) * MUL + m];
  else if (col <= 16) v = w4[(sp * 4 + (col - 13)) * MUL + m];
  else if (col <= 22) v = w5[(sp * 2 + (col - 17) / 3) * MUL + m];
  Wpack[tid] = v;
}

// ---------------- main kernel: monomials + WMMA GEMM + epilogue ---------
__launch_bounds__(MUL, 2)
__global__ void contract(const float* __restrict__ feats, const int* __restrict__ species,
                         const float* __restrict__ Mpair, const float* __restrict__ Wpack,
                         float* __restrict__ out) {
  extern __shared__ float smem[];
  float* Xs  = smem;                 // [MUL][KDIM] monomial rows (112.6 KB)
  float* Bls = Xs + MUL * KDIM;      // [224/2][NPAD][2] k-paired B (28 KB, TDM dest)
  float* Gs  = Bls + BLSF;           // [MUL][GSTR] g-values

  const int tid = threadIdx.x;
  const int n   = blockIdx.x;

  // ---- Tensor Data Mover: DMA the 28KB B-matrix tile global->LDS ------
  // D# per CDNA5 ISA: 1-D tensor, data_size=4B,
  // tensor_dim0 = tile_dim0 = 7168 elements, no multicast/pad/iterate.
  if (tid < 32) {                    // wave 0 issues one TDM op per workgroup
    unsigned int lds_addr =
        (unsigned int)(unsigned long long)(__attribute__((address_space(3))) float*)Bls;
    unsigned long long ga = (unsigned long long)Mpair;
    v4i g0;
    g0[0] = 1;                                           // count=1 (valid), user mode
    g0[1] = (int)lds_addr;                               // lds_addr
    g0[2] = (int)(unsigned int)ga;                       // global_addr[31:0]
    g0[3] = (int)((unsigned int)((ga >> 32) & 0x01FFFFFFu) | (2u << 30)); // type=2
    v8i g1;
    g1[0] = 0x00020000;                                  // wg_mask=0, data_size=2 (4B)
    g1[1] = (int)(BLSF << 16);                           // tensor_dim0[15:0] in [63:48]
    g1[2] = (int)(1u << 16);                             // tensor_dim0 hi=0, tensor_dim1=1
    g1[3] = (int)(BLSF << 16);                           // tile_dim0 = 7168
    g1[4] = 1;                                           // tile_dim1=1, tile_dim2=0
    g1[5] = BLSF;                                        // tensor_dim0_stride lo
    g1[6] = (int)(BLSF << 16);                           // stride hi=0, dim1_stride lo
    g1[7] = 0;                                           // dim1_stride hi
    asm volatile("tensor_load_to_lds %0, %1" :: "s"(g0), "s"(g1) : "memory");
  }

  // ---- gather this channel's 9 irrep components (0e | 1o | 2e layout) --
  const float* nf = feats + (size_t)n * (MUL * DIN);
  float f[9];
  f[0] = nf[tid];
  #pragma unroll
  for (int k = 0; k < 3; ++k) f[1 + k] = nf[MUL + tid * 3 + k];
  #pragma unroll
  for (int k = 0; k < 5; ++k) f[4 + k] = nf[MUL * 4 + tid * 5 + k];

  // prefetch the per-species epilogue weights while we compute/GEMM
  const int sp = species[n];
  const float* wp = Wpack + (size_t)sp * (WSTR * MUL);
  if (tid < NCOL) __builtin_prefetch(wp + tid * MUL, 0, 1);

  // ---- build symmetric monomial row in LDS (overlaps the TDM) ---------
  {
    float* xr = Xs + tid * KDIM;
    int idx = 0;
    #pragma unroll
    for (int a = 0; a < 9; ++a) xr[idx++] = f[a];
    #pragma unroll
    for (int a = 0; a < 9; ++a)
      #pragma unroll
      for (int b = a; b < 9; ++b) xr[idx++] = f[a] * f[b];
    #pragma unroll
    for (int a = 0; a < 9; ++a)
      #pragma unroll
      for (int b = a; b < 9; ++b) {
        float fab = f[a] * f[b];
        #pragma unroll
        for (int c = b; c < 9; ++c) xr[idx++] = fab * f[c];
      }
    xr[219] = 0.f;
  }

  if (tid < 32) __builtin_amdgcn_s_wait_tensorcnt(0);  // TDM tile landed
  __syncthreads();

  // ---- GEMM: each wave owns 2 M-tiles x 2 N-tiles, K=220 in steps of 4
  const int wave = tid >> 5;
  const int lane = tid & 31;
  const int l16  = lane & 15;
  const int lh   = lane >> 4;      // 0: K0/K1 half, 1: K2/K3 half

  v8f acc[2][2] = {};
  for (int k0 = 0; k0 < KDIM; k0 += 4) {
    const int pr = (k0 >> 1) + lh;
    v2f bf0 = *(const v2f*)&Bls[pr * (NPAD * 2) + l16 * 2];         // cols 0..15
    v2f bf1 = *(const v2f*)&Bls[pr * (NPAD * 2) + (16 + l16) * 2];  // cols 16..31
    #pragma unroll
    for (int mt = 0; mt < 2; ++mt) {
      const int m0 = (wave * 2 + mt) * 16;
      v2f af = *(const v2f*)&Xs[(m0 + l16) * KDIM + k0 + 2 * lh];
      acc[mt][0] = __builtin_amdgcn_wmma_f32_16x16x4_f32(
          false, af, false, bf0, (short)0, acc[mt][0], false, false);
      acc[mt][1] = __builtin_amdgcn_wmma_f32_16x16x4_f32(
          false, af, false, bf1, (short)0, acc[mt][1], false, false);
    }
  }

  // D layout: lane holds col = l16 (+16 per N-tile), rows lh*8 + r
  #pragma unroll
  for (int mt = 0; mt < 2; ++mt) {
    const int m0 = (wave * 2 + mt) * 16;
    #pragma unroll
    for (int nt = 0; nt < 2; ++nt) {
      const int col = nt * 16 + l16;
      if (col < NCOL) {
        #pragma unroll
        for (int r = 0; r < 8; ++r)
          Gs[(m0 + lh * 8 + r) * GSTR + col] = acc[mt][nt][r];
      }
    }
  }
  __syncthreads();

  // ---- epilogue: per-channel weighted reduction over the 23 g-functions
  const float* gr = Gs + tid * GSTR;
  float o0 = 0.f, o1 = 0.f, o2 = 0.f, o3 = 0.f;
  #pragma unroll
  for (int col = 0; col < NCOL; ++col) {
    float v = wp[col * MUL + tid] * gr[col];
    int p = kPtgt[col];
    if      (p == 0) o0 += v;
    else if (p == 1) o1 += v;
    else if (p == 2) o2 += v;
    else             o3 += v;
  }
  float* on = out + (size_t)n * 512;           // (mul*1 | mul*3) layout
  on[tid]               = o0;
  on[MUL + tid * 3 + 0] = o1;
  on[MUL + tid * 3 + 1] = o2;
  on[MUL + tid * 3 + 2] = o3;
}

extern "C" void kernel_launch(void* const* d_in, const int* in_sizes, int n_in,
                              void* d_out, int out_size, void* d_ws, size_t ws_size,
                              hipStream_t stream) {
  const float* feats   = (const float*)d_in[0];
  const int*   species = (const int*)  d_in[1];
  const float* b0 = (const float*)d_in[2];
  const float* w0 = (const float*)d_in[3];
  const float* b1 = (const float*)d_in[4];
  const float* w1 = (const float*)d_in[5];
  const float* b2 = (const float*)d_in[6];
  const float* w2 = (const float*)d_in[7];
  const float* b3 = (const float*)d_in[8];
  const float* w3 = (const float*)d_in[9];
  const float* b4 = (const float*)d_in[10];
  const float* w4 = (const float*)d_in[11];
  const float* b5 = (const float*)d_in[12];
  const float* w5 = (const float*)d_in[13];

  float* Mpair = (float*)d_ws;                 // 7168 floats (TDM tile)
  float* Wpack = Mpair + BLSF;                 // 10*24*128 floats

  build_M<<<(224 + 63) / 64, 64, 0, stream>>>(b0, b1, b2, b3, b4, b5, Mpair);
  build_W<<<(10 * WSTR * MUL + 255) / 256, 256, 0, stream>>>(w0, w1, w2, w3, w4, w5, Wpack);

  size_t shmem = (size_t)(MUL * KDIM + BLSF + MUL * GSTR) * sizeof(float);
  contract<<<NNODE, MUL, shmem, stream>>>(feats, species, Mpair, Wpack, (float*)d_out);
}